// Attention_54924041781953
// MI455X (gfx1250) — compile-verified
//
#include <hip/hip_runtime.h>
#include <hip/hip_bf16.h>
#include <math.h>

// ---------------------------------------------------------------------------
// 2D-RoPE multi-head attention, fp32, for MI455X (gfx1250, wave32).
// All matmuls on the matrix unit via V_WMMA_F32_16X16X4_F32.
//   b=8, n=1024 (32x32), c=1024, H=16, hd=64.
// Round-2 refinement: 2x2 register blocking per wave (4 accumulators),
// 128x64 block tiles -> 4 LDS b64 operand loads per 4 WMMAs (was 3 per 4).
// Kernel 1: QKV GEMM (8192x1024 @ 1024x3072) + fused 2D RoPE epilogue
// Kernel 2: flash attention, 128-row q blocks, streaming 64-row K/V tiles
// Kernel 3: output projection GEMM -> d_out
// Workspace: q,k,v,o buffers, 4 x 32MB = 128MB of d_ws.
// ---------------------------------------------------------------------------

typedef __attribute__((ext_vector_type(2))) float v2f;
typedef __attribute__((ext_vector_type(8))) float v8f;

#define LDT 68           // LDS row stride (floats): even (8B pairs) & 16B rows
#define BHD 64           // head dim
#define NTOK 1024
#define CDIM 1024
#define NHEADS 16
#define QSZ (8u*16u*1024u*64u)   // floats per q/k/v/o buffer

static __device__ inline v8f wmma4(v2f a, v2f b, v8f c) {
  // D = A(16x4 f32) x B(4x16 f32) + C(16x16 f32)
  return __builtin_amdgcn_wmma_f32_16x16x4_f32(
      /*neg_a=*/false, a, /*neg_b=*/false, b,
      /*c_mod=*/(short)0, c, /*reuse_a=*/false, /*reuse_b=*/false);
}

// Load a ROWSx64 fp32 tile (row stride ldg) into LDS with stride LDT.
// blockDim.x == 256; each thread moves ROWS/16 float4s.
template <int ROWS>
static __device__ inline void load_tile(const float* __restrict__ g, int ldg,
                                        float* __restrict__ lds, int tid) {
#pragma unroll
  for (int i = 0; i < ROWS / 16; ++i) {
    int idx = tid + i * 256;        // float4 slots
    int row = idx >> 4;             // 0..ROWS-1
    int c4  = idx & 15;             // 0..15
    float4 v = *(const float4*)(g + row * ldg + c4 * 4);
    *(float4*)(lds + row * LDT + c4 * 4) = v;
  }
}

// ---------------------------------------------------------------------------
// Kernel 1: QKV projection + 2D RoPE.
//   X: (8192, 1024) row-major, W: (3072, 1024) row-major.
//   grid = (64 M-blocks, 48 N-blocks), block = 256 (8 waves, 2x2 subtiles).
//   Each 64-wide N tile is exactly one of q/k/v for one head (192 = 3*64).
//   Outputs q,k,v in (b, H, n, 64) layout; RoPE fused on q,k.
// ---------------------------------------------------------------------------
__global__ __launch_bounds__(256) void qkv_rope_kernel(
    const float* __restrict__ X, const float* __restrict__ W,
    float* __restrict__ Q, float* __restrict__ K, float* __restrict__ V) {
  __shared__ float As[128 * LDT];
  __shared__ float Bs[64 * LDT];

  const int tid   = threadIdx.x;
  const int mBase = blockIdx.x * 128;  // token rows (0..8191)
  const int nBase = blockIdx.y * 64;   // qkv feature cols (0..3071)

  const int lane = tid & 31, half = lane >> 4, lr = lane & 15, w = tid >> 5;
  const int ti2 = (w >> 1) * 2, tj2 = (w & 1) * 2;   // 32x32 patch per wave

  v8f a00 = {}, a01 = {}, a10 = {}, a11 = {};

  for (int kt = 0; kt < CDIM; kt += 64) {
    load_tile<128>(X + (size_t)mBase * CDIM + kt, CDIM, As, tid);
    load_tile<64>(W + (size_t)nBase * CDIM + kt, CDIM, Bs, tid);
    __syncthreads();
#pragma unroll
    for (int kk = 0; kk < 64; kk += 4) {
      v2f ra0 = *(const v2f*)(As + ((ti2    ) * 16 + lr) * LDT + kk + 2 * half);
      v2f ra1 = *(const v2f*)(As + ((ti2 + 1) * 16 + lr) * LDT + kk + 2 * half);
      v2f rb0 = *(const v2f*)(Bs + ((tj2    ) * 16 + lr) * LDT + kk + 2 * half);
      v2f rb1 = *(const v2f*)(Bs + ((tj2 + 1) * 16 + lr) * LDT + kk + 2 * half);
      a00 = wmma4(ra0, rb0, a00);
      a01 = wmma4(ra0, rb1, a01);
      a10 = wmma4(ra1, rb0, a10);
      a11 = wmma4(ra1, rb1, a11);
    }
    __syncthreads();
  }

  // Stage the 128x64 result into LDS (reuse As) so RoPE can pair d with d+-32.
#pragma unroll
  for (int r = 0; r < 8; ++r) {
    int r0w = (ti2    ) * 16 + r + 8 * half;
    int r1w = (ti2 + 1) * 16 + r + 8 * half;
    As[r0w * LDT + (tj2    ) * 16 + lr] = a00[r];
    As[r0w * LDT + (tj2 + 1) * 16 + lr] = a01[r];
    As[r1w * LDT + (tj2    ) * 16 + lr] = a10[r];
    As[r1w * LDT + (tj2 + 1) * 16 + lr] = a11[r];
  }
  __syncthreads();

  const int h  = nBase / 192;
  const int r0 = nBase - h * 192;                 // 0 -> q, 64 -> k, 128 -> v
  float* dst = (r0 == 0) ? Q : (r0 == 64) ? K : V;

#pragma unroll
  for (int i = 0; i < 32; ++i) {
    int idx = tid + i * 256;                      // 0..8191
    int mm = idx >> 6, dd = idx & 63;
    int bt = mBase + mm;
    int p  = bt & (NTOK - 1);                     // token position in image
    int bI = bt >> 10;                            // batch index
    float val = As[mm * LDT + dd];
    if (r0 != 128) {                              // RoPE on q and k only
      float partner = As[mm * LDT + ((dd + 32) & 63)];
      float rot = (dd < 32) ? -partner : partner;
      int   j   = dd & 31;
      float pos = (dd < 32) ? (float)(p >> 5) : (float)(p & 31); // y : x
      // ifreq = 10000^(-j/32) = exp2(-j * log2(10000)/32)
      float theta = pos * exp2f((float)j * -0.41524101186092027f);
      float sn, cs;
      __sincosf(theta, &sn, &cs);
      val = val * cs + rot * sn;
    }
    dst[(((size_t)(bI * NHEADS + h) * NTOK + p) * BHD) + dd] = val;
  }
}

// ---------------------------------------------------------------------------
// Kernel 2: flash attention. grid = 128 (b*h) * 8 (q blocks) = 1024 blocks.
//   Per block: 128 q rows resident; stream 16 K/V tiles of 64 rows.
//   S = (QK^T)*0.125 via WMMA -> Ps in LDS -> online softmax -> O += P*V.
//   O written in (b, n, h, d) layout == (8192, 1024) row-major for proj GEMM.
// ---------------------------------------------------------------------------
__global__ __launch_bounds__(256) void flash_attn_kernel(
    const float* __restrict__ Q, const float* __restrict__ K,
    const float* __restrict__ V, float* __restrict__ O) {
  __shared__ float Qs[128 * LDT];
  __shared__ float Ks[64 * LDT];
  __shared__ float Vt[64 * LDT];   // transposed: Vt[d][kvRow]
  __shared__ float Ps[128 * LDT];
  __shared__ float row_m[128], row_l[128], row_s[128];

  const int tid = threadIdx.x;
  const int bh  = blockIdx.x >> 3;         // 0..127 = b*16+h
  const int qb  = blockIdx.x & 7;
  const int b   = bh >> 4, h = bh & 15;

  const int lane = tid & 31, half = lane >> 4, lr = lane & 15, w = tid >> 5;
  const int ti2 = (w >> 1) * 2, tj2 = (w & 1) * 2;

  load_tile<128>(Q + ((size_t)bh * NTOK + qb * 128) * BHD, BHD, Qs, tid);
  if (tid < 128) { row_m[tid] = -__builtin_inff(); row_l[tid] = 0.0f; }
  v8f o00 = {}, o01 = {}, o10 = {}, o11 = {};
  __syncthreads();

  for (int kv = 0; kv < 16; ++kv) {
    const float* kP = K + ((size_t)bh * NTOK + kv * 64) * BHD;
    const float* vP = V + ((size_t)bh * NTOK + kv * 64) * BHD;
    load_tile<64>(kP, BHD, Ks, tid);
#pragma unroll
    for (int i = 0; i < 4; ++i) {            // V tile, transposed into LDS
      int idx = tid + i * 256;
      int row = idx >> 4, c4 = idx & 15;
      float4 v = *(const float4*)(vP + row * BHD + c4 * 4);
      Vt[(c4 * 4 + 0) * LDT + row] = v.x;
      Vt[(c4 * 4 + 1) * LDT + row] = v.y;
      Vt[(c4 * 4 + 2) * LDT + row] = v.z;
      Vt[(c4 * 4 + 3) * LDT + row] = v.w;
    }
    __syncthreads();

    // S = Q * K^T (scaled by hd^-0.5 = 0.125)
    v8f s00 = {}, s01 = {}, s10 = {}, s11 = {};
#pragma unroll
    for (int kk = 0; kk < 64; kk += 4) {
      v2f ra0 = *(const v2f*)(Qs + ((ti2    ) * 16 + lr) * LDT + kk + 2 * half);
      v2f ra1 = *(const v2f*)(Qs + ((ti2 + 1) * 16 + lr) * LDT + kk + 2 * half);
      v2f rb0 = *(const v2f*)(Ks + ((tj2    ) * 16 + lr) * LDT + kk + 2 * half);
      v2f rb1 = *(const v2f*)(Ks + ((tj2 + 1) * 16 + lr) * LDT + kk + 2 * half);
      s00 = wmma4(ra0, rb0, s00);
      s01 = wmma4(ra0, rb1, s01);
      s10 = wmma4(ra1, rb0, s10);
      s11 = wmma4(ra1, rb1, s11);
    }
#pragma unroll
    for (int r = 0; r < 8; ++r) {
      int r0w = (ti2    ) * 16 + r + 8 * half;
      int r1w = (ti2 + 1) * 16 + r + 8 * half;
      Ps[r0w * LDT + (tj2    ) * 16 + lr] = s00[r] * 0.125f;
      Ps[r0w * LDT + (tj2 + 1) * 16 + lr] = s01[r] * 0.125f;
      Ps[r1w * LDT + (tj2    ) * 16 + lr] = s10[r] * 0.125f;
      Ps[r1w * LDT + (tj2 + 1) * 16 + lr] = s11[r] * 0.125f;
    }
    __syncthreads();

    // Online softmax: one thread per q row.
    if (tid < 128) {
      float mOld = row_m[tid];
      float mNew = mOld;
#pragma unroll 8
      for (int j = 0; j < 64; ++j) mNew = fmaxf(mNew, Ps[tid * LDT + j]);
      float sc  = __expf(mOld - mNew);       // exp(-inf)=0 on first tile
      float sum = 0.0f;
#pragma unroll 8
      for (int j = 0; j < 64; ++j) {
        float pj = __expf(Ps[tid * LDT + j] - mNew);
        Ps[tid * LDT + j] = pj;
        sum += pj;
      }
      row_m[tid] = mNew;
      row_l[tid] = row_l[tid] * sc + sum;
      row_s[tid] = sc;
    }
    __syncthreads();

    // Rescale running O, then O += P * V.
#pragma unroll
    for (int r = 0; r < 8; ++r) {
      float f0 = row_s[(ti2    ) * 16 + r + 8 * half];
      float f1 = row_s[(ti2 + 1) * 16 + r + 8 * half];
      o00[r] *= f0; o01[r] *= f0;
      o10[r] *= f1; o11[r] *= f1;
    }
#pragma unroll
    for (int kk = 0; kk < 64; kk += 4) {
      v2f ra0 = *(const v2f*)(Ps + ((ti2    ) * 16 + lr) * LDT + kk + 2 * half);
      v2f ra1 = *(const v2f*)(Ps + ((ti2 + 1) * 16 + lr) * LDT + kk + 2 * half);
      v2f rb0 = *(const v2f*)(Vt + ((tj2    ) * 16 + lr) * LDT + kk + 2 * half);
      v2f rb1 = *(const v2f*)(Vt + ((tj2 + 1) * 16 + lr) * LDT + kk + 2 * half);
      o00 = wmma4(ra0, rb0, o00);
      o01 = wmma4(ra0, rb1, o01);
      o10 = wmma4(ra1, rb0, o10);
      o11 = wmma4(ra1, rb1, o11);
    }
    __syncthreads();                          // protect Ks/Vt/Ps for next tile
  }

  // Normalize by l and store in (b, n, h, d) layout.
#pragma unroll
  for (int r = 0; r < 8; ++r) {
#pragma unroll
    for (int i = 0; i < 2; ++i) {
      int rowIn = (ti2 + i) * 16 + r + 8 * half;
      float inv = 1.0f / row_l[rowIn];
      int p  = qb * 128 + rowIn;
      int d0 = tj2 * 16 + lr;
      size_t base = (((size_t)(b * NTOK + p) * NHEADS + h) * BHD);
      v8f* oa = i ? &o10 : &o00;
      v8f* ob = i ? &o11 : &o01;
      O[base + d0]      = (*oa)[r] * inv;
      O[base + d0 + 16] = (*ob)[r] * inv;
    }
  }
}

// ---------------------------------------------------------------------------
// Kernel 3: output projection. out(8192x1024) = O(8192x1024) @ Wp^T.
//   grid = (64, 16), block = 256.
// ---------------------------------------------------------------------------
__global__ __launch_bounds__(256) void proj_kernel(
    const float* __restrict__ Oin, const float* __restrict__ W,
    float* __restrict__ out) {
  __shared__ float As[128 * LDT];
  __shared__ float Bs[64 * LDT];

  const int tid   = threadIdx.x;
  const int mBase = blockIdx.x * 128;
  const int nBase = blockIdx.y * 64;

  const int lane = tid & 31, half = lane >> 4, lr = lane & 15, w = tid >> 5;
  const int ti2 = (w >> 1) * 2, tj2 = (w & 1) * 2;

  v8f a00 = {}, a01 = {}, a10 = {}, a11 = {};

  for (int kt = 0; kt < CDIM; kt += 64) {
    load_tile<128>(Oin + (size_t)mBase * CDIM + kt, CDIM, As, tid);
    load_tile<64>(W + (size_t)nBase * CDIM + kt, CDIM, Bs, tid);
    __syncthreads();
#pragma unroll
    for (int kk = 0; kk < 64; kk += 4) {
      v2f ra0 = *(const v2f*)(As + ((ti2    ) * 16 + lr) * LDT + kk + 2 * half);
      v2f ra1 = *(const v2f*)(As + ((ti2 + 1) * 16 + lr) * LDT + kk + 2 * half);
      v2f rb0 = *(const v2f*)(Bs + ((tj2    ) * 16 + lr) * LDT + kk + 2 * half);
      v2f rb1 = *(const v2f*)(Bs + ((tj2 + 1) * 16 + lr) * LDT + kk + 2 * half);
      a00 = wmma4(ra0, rb0, a00);
      a01 = wmma4(ra0, rb1, a01);
      a10 = wmma4(ra1, rb0, a10);
      a11 = wmma4(ra1, rb1, a11);
    }
    __syncthreads();
  }

#pragma unroll
  for (int r = 0; r < 8; ++r) {
    int row0 = mBase + (ti2    ) * 16 + r + 8 * half;
    int row1 = mBase + (ti2 + 1) * 16 + r + 8 * half;
    out[(size_t)row0 * CDIM + nBase + (tj2    ) * 16 + lr] = a00[r];
    out[(size_t)row0 * CDIM + nBase + (tj2 + 1) * 16 + lr] = a01[r];
    out[(size_t)row1 * CDIM + nBase + (tj2    ) * 16 + lr] = a10[r];
    out[(size_t)row1 * CDIM + nBase + (tj2 + 1) * 16 + lr] = a11[r];
  }
}

// ---------------------------------------------------------------------------
extern "C" void kernel_launch(void* const* d_in, const int* in_sizes, int n_in,
                              void* d_out, int out_size, void* d_ws, size_t ws_size,
                              hipStream_t stream) {
  (void)in_sizes; (void)n_in; (void)out_size; (void)ws_size;
  const float* x     = (const float*)d_in[0];   // (8, 1024, 1024)
  const float* wqkv  = (const float*)d_in[1];   // (3072, 1024)
  const float* wproj = (const float*)d_in[2];   // (1024, 1024)
  float* out = (float*)d_out;                   // (8, 1024, 1024)

  float* ws = (float*)d_ws;                     // needs 4*QSZ floats = 128 MB
  float* q = ws;
  float* k = ws + (size_t)QSZ;
  float* v = ws + (size_t)2 * QSZ;
  float* o = ws + (size_t)3 * QSZ;

  qkv_rope_kernel<<<dim3(64, 48), 256, 0, stream>>>(x, wqkv, q, k, v);
  flash_attn_kernel<<<dim3(1024), 256, 0, stream>>>(q, k, v, o);
  proj_kernel<<<dim3(64, 16), 256, 0, stream>>>(o, wproj, out);
}